// Transform_40003325395637
// MI455X (gfx1250) — compile-verified
//
#include <hip/hip_runtime.h>
#include <hip/hip_bf16.h>

typedef __attribute__((ext_vector_type(2))) float v2f;
typedef __attribute__((ext_vector_type(8))) float v8f;

#define TRI_EPS 1e-12f

// ---------------- Kernel 1: row squared norms (one wave per row) ----------------
__global__ __launch_bounds__(256) void row_sqnorm_kernel(
    const float* __restrict__ X, float* __restrict__ sq, int n, int d) {
  const int lane = threadIdx.x & 31;
  const int wid  = threadIdx.x >> 5;
  const int row  = blockIdx.x * 8 + wid;
  if (row >= n) return;
  const float* xr = X + (size_t)row * d;
  float s = 0.0f;
  for (int c = lane; c < d; c += 32) {
    float v = xr[c];
    s += v * v;
  }
  #pragma unroll
  for (int off = 16; off > 0; off >>= 1) s += __shfl_xor(s, off, 32);
  if (lane == 0) sq[row] = s;
}

// Per-lane mining of one 16x16 Gram tile: compare clamped SQUARED distances
// (sqrt is monotonic and only indices are consumed, so it can be skipped).
// Each lane updates a running best over the columns it owns; no cross-lane
// traffic here — the 16-lane arg-reduction happens once at the end.
__device__ __forceinline__ void mine_tile_local(
    const v8f& acc, int col, const int* __restrict__ tgt,
    const float* __restrict__ sq, const int* tm, const float* sm,
    float* bpV, int* bpI, float* bnV, int* bnI) {
  const int   tn  = tgt[col];
  const float sqn = sq[col];
  #pragma unroll
  for (int r = 0; r < 8; ++r) {
    float d2   = fmaxf(sm[r] + sqn - 2.0f * acc[r], TRI_EPS);
    bool  same = (tm[r] == tn);
    float pv = same ? d2 : -__builtin_inff();
    float nv = same ?  __builtin_inff() : d2;
    // columns visited by a lane ascend monotonically -> strict compare keeps
    // the first occurrence (matches jnp.argmax/argmin tie-breaking).
    if (pv > bpV[r]) { bpV[r] = pv; bpI[r] = col; }
    if (nv < bnV[r]) { bnV[r] = nv; bnI[r] = col; }
  }
}

// ---------------- Kernel 2: WMMA Gram tiles + batch-hard mining + output copies ----
// Block = 256 threads = 8 waves. Block owns a 16-row band (m0..m0+15).
// Each wave computes FOUR 16x16 Gram tiles at a time (64 columns) via
// V_WMMA_F32_16X16X4_F32, sharing one A fragment across 4 accumulators
// (5 loads : 4 WMMAs per K-step, 4 independent accumulator chains).
__global__ __launch_bounds__(256) void triplet_mine_kernel(
    const float* __restrict__ X, const int* __restrict__ tgt,
    const float* __restrict__ sq, float* __restrict__ out, int n, int d) {
  __builtin_assume(d % 64 == 0 && d > 0);
  __builtin_assume(n % 64 == 0 && n > 0);

  const int tid  = threadIdx.x;
  const int lane = tid & 31;
  const int wid  = tid >> 5;
  const int m0   = blockIdx.x * 16;
  const int hi   = lane >> 4;          // 0: lanes 0-15, 1: lanes 16-31
  const int l15  = lane & 15;
  const int koff = hi * 2;             // A/B fragment K sub-offset per ISA layout

  // Targets / norms for the 8 C-rows this lane owns (M = r + 8*hi).
  int   tm[8];
  float sm[8];
  #pragma unroll
  for (int r = 0; r < 8; ++r) {
    tm[r] = tgt[m0 + r + 8 * hi];
    sm[r] = sq [m0 + r + 8 * hi];
  }

  const float* Arow = X + (size_t)(m0 + l15) * d + koff;

  float bpV[8], bnV[8];
  int   bpI[8], bnI[8];
  #pragma unroll
  for (int r = 0; r < 8; ++r) {
    bpV[r] = -__builtin_inff(); bpI[r] = 0;
    bnV[r] =  __builtin_inff(); bnI[r] = 0;
  }

  const int groups = n >> 6;           // groups of 4 column tiles (64 cols)
  for (int g = wid; g < groups; g += 8) {
    const int n0 = g << 6;
    const float* B0 = X + (size_t)(n0 + l15) * d + koff;
    const float* B1 = B0 + (size_t)16 * d;
    const float* B2 = B0 + (size_t)32 * d;
    const float* B3 = B0 + (size_t)48 * d;

    v8f acc0 = {}, acc1 = {}, acc2 = {}, acc3 = {};
    #pragma unroll 2
    for (int k0 = 0; k0 < d; k0 += 4) {
      v2f a  = *(const v2f*)(Arow + k0);
      v2f b0 = *(const v2f*)(B0 + k0);
      v2f b1 = *(const v2f*)(B1 + k0);
      v2f b2 = *(const v2f*)(B2 + k0);
      v2f b3 = *(const v2f*)(B3 + k0);
      acc0 = __builtin_amdgcn_wmma_f32_16x16x4_f32(false, a, false, b0, (short)0, acc0, false, false);
      acc1 = __builtin_amdgcn_wmma_f32_16x16x4_f32(false, a, false, b1, (short)0, acc1, false, false);
      acc2 = __builtin_amdgcn_wmma_f32_16x16x4_f32(false, a, false, b2, (short)0, acc2, false, false);
      acc3 = __builtin_amdgcn_wmma_f32_16x16x4_f32(false, a, false, b3, (short)0, acc3, false, false);
    }

    mine_tile_local(acc0, n0      + l15, tgt, sq, tm, sm, bpV, bpI, bnV, bnI);
    mine_tile_local(acc1, n0 + 16 + l15, tgt, sq, tm, sm, bpV, bpI, bnV, bnI);
    mine_tile_local(acc2, n0 + 32 + l15, tgt, sq, tm, sm, bpV, bpI, bnV, bnI);
    mine_tile_local(acc3, n0 + 48 + l15, tgt, sq, tm, sm, bpV, bpI, bnV, bnI);
  }

  // -------- one 16-lane arg-reduction per row (once per wave) --------
  #pragma unroll
  for (int r = 0; r < 8; ++r) {
    float pv = bpV[r]; int pi = bpI[r];
    float nv = bnV[r]; int ni = bnI[r];
    #pragma unroll
    for (int s = 1; s < 16; s <<= 1) {
      float opv = __shfl_xor(pv, s, 16);
      int   opi = __shfl_xor(pi, s, 16);
      if (opv > pv || (opv == pv && opi < pi)) { pv = opv; pi = opi; }
      float onv = __shfl_xor(nv, s, 16);
      int   oni = __shfl_xor(ni, s, 16);
      if (onv < nv || (onv == nv && oni < ni)) { nv = onv; ni = oni; }
    }
    bpV[r] = pv; bpI[r] = pi;
    bnV[r] = nv; bnI[r] = ni;
  }

  // -------- cross-wave merge via LDS --------
  __shared__ float sPV[8][16], sNV[8][16];
  __shared__ int   sPI[8][16], sNI[8][16];
  __shared__ int   fPos[16], fNeg[16];

  if (l15 == 0) {
    #pragma unroll
    for (int r = 0; r < 8; ++r) {
      int row = r + 8 * hi;
      sPV[wid][row] = bpV[r]; sPI[wid][row] = bpI[r];
      sNV[wid][row] = bnV[r]; sNI[wid][row] = bnI[r];
    }
  }
  __syncthreads();

  if (tid < 16) {
    float pv = -__builtin_inff(); int pi = 0;
    float nv =  __builtin_inff(); int ni = 0;
    for (int w = 0; w < 8; ++w) {
      float v = sPV[w][tid]; int i = sPI[w][tid];
      if (v > pv || (v == pv && i < pi)) { pv = v; pi = i; }
      v = sNV[w][tid]; i = sNI[w][tid];
      if (v < nv || (v == nv && i < ni)) { nv = v; ni = i; }
    }
    fPos[tid] = pi;
    fNeg[tid] = ni;
  }
  __syncthreads();

  // -------- output copies: pair1 (X twice), pair2 (gather pos/neg), y --------
  const int vpr = d >> 2;                  // float4 per row
  const size_t nd4 = (size_t)n * vpr;
  const float4* X4 = (const float4*)X;
  float4* O4 = (float4*)out;
  const int total = 4 * 16 * vpr;
  for (int idx = tid; idx < total; idx += 256) {
    int seg = idx / (16 * vpr);
    int rem = idx - seg * 16 * vpr;
    int r   = rem / vpr;
    int c   = rem - r * vpr;
    int srcRow = (seg < 2) ? (m0 + r) : (seg == 2 ? fPos[r] : fNeg[r]);
    O4[(size_t)seg * nd4 + (size_t)(m0 + r) * vpr + c] =
        X4[(size_t)srcRow * vpr + c];
  }
  float* y = out + (size_t)4 * n * d;
  if (tid < 16)       y[m0 + tid] = 1.0f;
  else if (tid < 32)  y[(size_t)n + m0 + (tid - 16)] = 0.0f;
}

extern "C" void kernel_launch(void* const* d_in, const int* in_sizes, int n_in,
                              void* d_out, int out_size, void* d_ws, size_t ws_size,
                              hipStream_t stream) {
  const float* X   = (const float*)d_in[0];
  const int*   tgt = (const int*)d_in[1];
  const int n = in_sizes[1];          // 4096 targets
  const int d = in_sizes[0] / n;      // 2048
  float* sq  = (float*)d_ws;          // n floats of scratch
  float* out = (float*)d_out;

  row_sqnorm_kernel<<<(n + 7) / 8, 256, 0, stream>>>(X, sq, n, d);
  triplet_mine_kernel<<<n / 16, 256, 0, stream>>>(X, tgt, sq, out, n, d);
}